// History_61692910240163
// MI455X (gfx1250) — compile-verified
//
#include <hip/hip_runtime.h>

// ---------------------------------------------------------------------------
// History.push ring-buffer scatter for MI455X (gfx1250, wave32).
//
// Row norms are computed on the matrix pipe: V_WMMA_F32_16X16X4_F32 with
// A == B (the 16x4 f32 A-layout is its own transpose-compatible B-layout),
// giving D = G * G^T; diag(D) = squared row norms, full f32 precision.
// Everything else is stream-ordered memory movement (B128 copies, scans,
// conflict-free scatters) sized for 23.3 TB/s HBM.
// ---------------------------------------------------------------------------

typedef float v2f __attribute__((ext_vector_type(2)));
typedef float v8f __attribute__((ext_vector_type(8)));

#define DIM 128
#define CHUNK 1024         // elements scanned per block (256 thr x 4)

// ---------- init copies (d_out must start as a copy of the inputs) ----------
__global__ void copy_f4(const float4* __restrict__ in, float4* __restrict__ out, int n) {
    int i = blockIdx.x * blockDim.x + threadIdx.x;
    if (i < n) out[i] = in[i];
}

__global__ void copy_i2f(const int* __restrict__ in, float* __restrict__ out, int n) {
    int i = blockIdx.x * blockDim.x + threadIdx.x;
    if (i < n) out[i] = (float)in[i];
}

// ---------- row norms via WMMA: one wave per 16-row tile ----------
__global__ __launch_bounds__(256)
void norms_wmma(const float* __restrict__ grad, const float* __restrict__ thresh_p,
                int* __restrict__ keep, int* __restrict__ del, int B) {
    const int lane = threadIdx.x & 31;
    const int wave = threadIdx.x >> 5;
    const int tile = blockIdx.x * 8 + wave;        // 16 rows per tile
    const int row0 = tile * 16;
    if (row0 >= B) return;                          // uniform per wave

    const float t  = thresh_p[0];
    const float t2 = t * t;

    // A-matrix 16x4 f32 layout: lane L holds row M = L&15;
    // VGPR0/1 = K{0,1} (lanes 0-15) or K{2,3} (lanes 16-31).
    const int m  = lane & 15;
    const int kh = (lane >> 4) * 2;                 // 0 or 2
    const float* base = grad + (size_t)(row0 + m) * DIM + kh;

    v8f c = {};
#pragma unroll
    for (int k0 = 0; k0 < DIM; k0 += 4) {
        v2f a = *(const v2f*)(base + k0);
        // D = A * A^T + C  (same registers for A and B: symmetric layout)
        c = __builtin_amdgcn_wmma_f32_16x16x4_f32(
                false, a, false, a, (short)0, c, false, false);
    }

    // diag(D): row r lives in VGPR r at lane r (r<8) and lane 24+r (row 8+r)
    float d = 0.f;
#pragma unroll
    for (int r = 0; r < 8; ++r)
        if (lane == r || lane == 24 + r) d = c[r];

    if (lane < 8 || lane >= 24) {
        int row = row0 + ((lane < 8) ? lane : (lane - 16));
        keep[row] = (d <= t2) ? 1 : 0;    // gnorm <= thresh
        del[row]  = (d >= t2) ? 1 : 0;    // gnorm >= thresh
    }
}

// ---------- hierarchical scan of keep flags ----------
__global__ void chunk_sums(const int* __restrict__ keep, int* __restrict__ csum, int B) {
    __shared__ int s[256];
    const int t = threadIdx.x;
    const int base = blockIdx.x * CHUNK + t * 4;
    int v = 0;
    if (base + 3 < B) v = keep[base] + keep[base+1] + keep[base+2] + keep[base+3];
    s[t] = v;
    __syncthreads();
    for (int off = 128; off > 0; off >>= 1) {
        if (t < off) s[t] += s[t + off];
        __syncthreads();
    }
    if (t == 0) csum[blockIdx.x] = s[0];
}

__global__ void scan_chunks(const int* __restrict__ csum, int* __restrict__ coff, int nch,
                            const int* __restrict__ index_p, float* __restrict__ new_index_out,
                            int E) {
    __shared__ int s[256];
    const int t = threadIdx.x;
    const int v = (t < nch) ? csum[t] : 0;
    s[t] = v;
    __syncthreads();
    for (int off = 1; off < 256; off <<= 1) {       // Hillis-Steele inclusive
        int x = (t >= off) ? s[t - off] : 0;
        __syncthreads();
        s[t] += x;
        __syncthreads();
    }
    if (t < nch) coff[t] = s[t] - v;                // exclusive prefix
    if (t == 255) {
        int total = s[255];
        int idx0  = index_p[0] % E;
        *new_index_out = (float)(idx0 + total);     // new_index (not wrapped)
    }
}

__global__ void compute_slots(const int* __restrict__ keep, const int* __restrict__ coff,
                              int* __restrict__ slots, const int* __restrict__ index_p,
                              int B, int E) {
    __shared__ int s[256];
    const int t = threadIdx.x;
    const int base = blockIdx.x * CHUNK + t * 4;
    int k0 = 0, k1 = 0, k2 = 0, k3 = 0;
    if (base + 3 < B) { k0 = keep[base]; k1 = keep[base+1]; k2 = keep[base+2]; k3 = keep[base+3]; }
    const int tot = k0 + k1 + k2 + k3;
    s[t] = tot;
    __syncthreads();
    for (int off = 1; off < 256; off <<= 1) {
        int x = (t >= off) ? s[t - off] : 0;
        __syncthreads();
        s[t] += x;
        __syncthreads();
    }
    int c = s[t] - tot + coff[blockIdx.x];          // exclusive prefix incl. chunk offset
    const int idx0 = index_p[0] % E;
    if (base + 3 < B) {
        int* so = slots + base;
        c += k0; if (k0) so[0] = (idx0 + c - 1) % E;
        c += k1; if (k1) so[1] = (idx0 + c - 1) % E;
        c += k2; if (k2) so[2] = (idx0 + c - 1) % E;
        c += k3; if (k3) so[3] = (idx0 + c - 1) % E;
    }
}

// ---------- phase 1: evict high-grad cached entries (reads ORIGINAL pos) ----------
__global__ void del_scatter(const int* __restrict__ del, const int* __restrict__ gids,
                            const int* __restrict__ pos_in,
                            float* __restrict__ out_pos, float* __restrict__ out_itg,
                            int B, int E) {
    int i = blockIdx.x * blockDim.x + threadIdx.x;
    if (i >= B || !del[i]) return;
    const int g = gids[i];
    out_pos[g] = -1.0f;
    int di = pos_in[g];
    if (di < 0) di += E;          // torch index wrap: -1 -> E-1 (yes, even uncached)
    out_itg[di] = -1.0f;
}

// ---------- phase 2: evict current occupants of the assigned slots ----------
__global__ void evict_scatter(const int* __restrict__ keep, const int* __restrict__ slots,
                              const float* __restrict__ out_itg, float* __restrict__ out_pos,
                              int B) {
    int i = blockIdx.x * blockDim.x + threadIdx.x;
    if (i >= B || !keep[i]) return;
    const int eg = (int)out_itg[slots[i]];          // read AFTER del writes
    if (eg != -1) out_pos[eg] = -1.0f;
}

// ---------- phase 3: final writes (win all conflicts) ----------
__global__ __launch_bounds__(256)
void write_rows(const int* __restrict__ keep, const int* __restrict__ slots,
                const int* __restrict__ gids, const float* __restrict__ feats,
                float* __restrict__ out_emb, float* __restrict__ out_pos,
                float* __restrict__ out_itg, int B) {
    const int lane = threadIdx.x & 31;
    const int row  = blockIdx.x * 8 + (threadIdx.x >> 5);   // one wave per row
    if (row >= B || !keep[row]) return;
    const int s = slots[row];
    const float4* src = (const float4*)(feats + (size_t)row * DIM);
    float4*       dst = (float4*)(out_emb + (size_t)s * DIM);
    dst[lane] = src[lane];                                  // 32 lanes x 16B = 512B row
    if (lane == 0) {
        const int g = gids[row];
        out_pos[g] = (float)s;
        out_itg[s] = (float)g;
    }
}

extern "C" void kernel_launch(void* const* d_in, const int* in_sizes, int n_in,
                              void* d_out, int out_size, void* d_ws, size_t ws_size,
                              hipStream_t stream) {
    const float* emb_in  = (const float*)d_in[0];
    const int*   pos_in  = (const int*)  d_in[1];
    const int*   itg_in  = (const int*)  d_in[2];
    const int*   gids    = (const int*)  d_in[3];
    const float* feats   = (const float*)d_in[4];
    const float* grad    = (const float*)d_in[5];
    const float* thresh  = (const float*)d_in[6];
    const int*   index_p = (const int*)  d_in[7];

    const int N = in_sizes[1];          // 2,000,000 nodes
    const int E = in_sizes[2];          // 800,000 ring slots
    const int B = in_sizes[3];          // 262,144 batch rows
    const int NCH = (B + CHUNK - 1) / CHUNK;   // 256 scan chunks

    // outputs: [emb | pos | index_to_gid | new_index], ints encoded as f32
    float* out_emb = (float*)d_out;
    float* out_pos = out_emb + (size_t)E * DIM;
    float* out_itg = out_pos + (size_t)N;
    float* out_ni  = out_itg + (size_t)E;

    // workspace ints: keep[B] del[B] slots[B] csum[NCH] coff[NCH]  (~3.1 MB)
    int* ws    = (int*)d_ws;
    int* keep  = ws;
    int* del   = ws + (size_t)B;
    int* slots = ws + (size_t)2 * B;
    int* csum  = ws + (size_t)3 * B;
    int* coff  = csum + NCH;

    const int TB = 256;

    // init copies of the three state arrays into d_out
    {
        int n4 = E * DIM / 4;
        copy_f4<<<(n4 + TB - 1) / TB, TB, 0, stream>>>((const float4*)emb_in, (float4*)out_emb, n4);
        copy_i2f<<<(N + TB - 1) / TB, TB, 0, stream>>>(pos_in, out_pos, N);
        copy_i2f<<<(E + TB - 1) / TB, TB, 0, stream>>>(itg_in, out_itg, E);
    }

    // row norms on the matrix pipe (8 waves -> 8 tiles of 16 rows per block)
    {
        int tiles = B / 16;
        norms_wmma<<<(tiles + 7) / 8, TB, 0, stream>>>(grad, thresh, keep, del, B);
    }

    // scan keep -> ring slots; also emits new_index
    chunk_sums   <<<NCH, TB, 0, stream>>>(keep, csum, B);
    scan_chunks  <<<1,   TB, 0, stream>>>(csum, coff, NCH, index_p, out_ni, E);
    compute_slots<<<NCH, TB, 0, stream>>>(keep, coff, slots, index_p, B, E);

    // ordered scatter phases (stream order == reference update order)
    del_scatter  <<<(B + TB - 1) / TB, TB, 0, stream>>>(del, gids, pos_in, out_pos, out_itg, B, E);
    evict_scatter<<<(B + TB - 1) / TB, TB, 0, stream>>>(keep, slots, out_itg, out_pos, B);
    write_rows   <<<(B + 7) / 8,       TB, 0, stream>>>(keep, slots, gids, feats,
                                                        out_emb, out_pos, out_itg, B);
}